// Clustering_36318243455201
// MI455X (gfx1250) — compile-verified
//
#include <hip/hip_runtime.h>
#include <math.h>

typedef __attribute__((ext_vector_type(2))) float v2f;
typedef __attribute__((ext_vector_type(8))) float v8f;

#define N_ROWS 16384
#define K_CLUST 1024
#define D_DIM 256
#define BK 64
#define BSTR 68   // 64 + 4 pad: stride%64 == 4 -> conflict-free fragment reads

// ---------------- row sum-of-squares: one wave per row ----------------
__global__ __launch_bounds__(256)
void rowsq_kernel(const float* __restrict__ m, float* __restrict__ out, int rows) {
    int wave = threadIdx.x >> 5;
    int lane = threadIdx.x & 31;
    int row  = blockIdx.x * 8 + wave;
    if (row >= rows) return;
    const float* p = m + (size_t)row * D_DIM + lane * 8;
    float4 a = *(const float4*)p;
    float4 b = *(const float4*)(p + 4);
    float s = a.x*a.x + a.y*a.y + a.z*a.z + a.w*a.w
            + b.x*b.x + b.y*b.y + b.z*b.z + b.w*b.w;
    for (int off = 16; off > 0; off >>= 1) s += __shfl_xor(s, off, 32);
    if (lane == 0) out[row] = s;
}

// ---------------- fused GEMM + Student-t kernel + partial sum ----------------
// Block tile: 128 rows x 128 cols. 8 waves; wave w owns rows [w*16, w*16+16).
// cross = A(16x256) x W^T -> per wave 8 WMMA tiles along cols.
__global__ __launch_bounds__(256)
void qgemm_kernel(const float* __restrict__ A,      // (N, D) comz
                  const float* __restrict__ W,      // (K, D) weights
                  const float* __restrict__ z2,     // (N)
                  const float* __restrict__ w2,     // (K)
                  float* __restrict__ qbuf,         // (N, K) q1 output (scratch in d_out)
                  float* __restrict__ partials) {   // (gridDim.y*gridDim.x)
    __shared__ float bt[128 * BSTR];
    __shared__ float wsum[8];

    const int tid  = threadIdx.x;
    const int wave = tid >> 5;
    const int lane = tid & 31;
    const int lo   = lane & 15;
    const int hi   = lane >> 4;            // 0 or 1
    const int sel  = hi * 2;               // K sub-select per ISA 16x4 f32 layout
    const int col0 = blockIdx.x * 128;
    const int row0 = blockIdx.y * 128;
    const int rowBase = row0 + wave * 16;

    v8f acc[8];
    #pragma unroll
    for (int t = 0; t < 8; ++t)
        #pragma unroll
        for (int j = 0; j < 8; ++j) acc[t][j] = 0.0f;

    // A fragment base: lane holds row (rowBase+lo), K pair starting at k+sel
    const float* aptr = A + (size_t)(rowBase + lo) * D_DIM + sel;

    for (int kc = 0; kc < D_DIM; kc += BK) {
        __syncthreads();
        // Cooperative load: W tile (128 cols x 64 k) -> LDS, coalesced float4
        #pragma unroll
        for (int i = 0; i < 8; ++i) {
            int idx = tid + i * 256;       // 0..2047 float4 slots
            int c   = idx >> 4;            // 16 float4 per col-row
            int kq  = idx & 15;
            float4 v = *(const float4*)(W + (size_t)(col0 + c) * D_DIM + kc + kq * 4);
            *(float4*)&bt[c * BSTR + kq * 4] = v;
        }
        __syncthreads();

        #pragma unroll
        for (int kk = 0; kk < BK; kk += 4) {
            v2f a = *(const v2f*)(aptr + kc + kk);
            #pragma unroll
            for (int t = 0; t < 8; ++t) {
                // B frag: lane lo -> col (t*16+lo); V0/V1 = K (kk+sel, kk+sel+1)
                v2f b = *(const v2f*)&bt[(t * 16 + lo) * BSTR + kk + sel];
                acc[t] = __builtin_amdgcn_wmma_f32_16x16x4_f32(
                    false, a, false, b, (short)0, acc[t], false, false);
            }
        }
    }

    // Epilogue: d2 = max(z2 + w2 - 2*cross, 0); q1 = 1/(1+d2)
    float zr[8];
    #pragma unroll
    for (int r = 0; r < 8; ++r) zr[r] = z2[rowBase + r + 8 * hi];

    float lsum = 0.0f;
    #pragma unroll
    for (int t = 0; t < 8; ++t) {
        int col  = col0 + t * 16 + lo;
        float wv = w2[col];
        #pragma unroll
        for (int r = 0; r < 8; ++r) {
            float cross = acc[t][r];       // element (M=r+8*hi, N=lo)
            float d2 = fmaxf(zr[r] + wv - 2.0f * cross, 0.0f);
            float q1 = 1.0f / (1.0f + d2);
            qbuf[(size_t)(rowBase + r + 8 * hi) * K_CLUST + col] = q1;
            lsum += q1;
        }
    }
    // Deterministic block reduction (shfl tree + fixed-order LDS fold)
    for (int off = 16; off > 0; off >>= 1) lsum += __shfl_xor(lsum, off, 32);
    if (lane == 0) wsum[wave] = lsum;
    __syncthreads();
    if (tid == 0) {
        float s = 0.0f;
        #pragma unroll
        for (int i = 0; i < 8; ++i) s += wsum[i];
        partials[blockIdx.y * gridDim.x + blockIdx.x] = s;
    }
}

// ---------------- fold 1024 block partials -> scalar S ----------------
__global__ __launch_bounds__(256)
void reduce_kernel(const float* __restrict__ partials, float* __restrict__ S) {
    __shared__ float wsum[8];
    int t = threadIdx.x;
    float s = partials[t] + partials[t + 256] + partials[t + 512] + partials[t + 768];
    for (int off = 16; off > 0; off >>= 1) s += __shfl_xor(s, off, 32);
    if ((t & 31) == 0) wsum[t >> 5] = s;
    __syncthreads();
    if (t == 0) {
        float tot = 0.0f;
        for (int i = 0; i < 8; ++i) tot += wsum[i];
        S[0] = tot;
    }
}

// ---------------- q = q1/S (in place), loss = log(q) ----------------
__global__ __launch_bounds__(256)
void finalize_kernel(float* __restrict__ loss, float* __restrict__ q,
                     const float* __restrict__ S) {
    const float invS = 1.0f / S[0];
    const size_t nv = (size_t)N_ROWS * K_CLUST / 4;
    size_t i = (size_t)blockIdx.x * blockDim.x + threadIdx.x;
    for (; i < nv; i += (size_t)gridDim.x * blockDim.x) {
        float4 v = ((const float4*)q)[i];
        float4 qq, lg;
        qq.x = v.x * invS; qq.y = v.y * invS; qq.z = v.z * invS; qq.w = v.w * invS;
        lg.x = logf(qq.x); lg.y = logf(qq.y); lg.z = logf(qq.z); lg.w = logf(qq.w);
        ((float4*)q)[i]    = qq;
        ((float4*)loss)[i] = lg;
    }
}

extern "C" void kernel_launch(void* const* d_in, const int* in_sizes, int n_in,
                              void* d_out, int out_size, void* d_ws, size_t ws_size,
                              hipStream_t stream) {
    (void)in_sizes; (void)n_in; (void)out_size; (void)ws_size;
    const float* comz    = (const float*)d_in[0];   // (16384, 256) f32
    const float* weights = (const float*)d_in[1];   // (1024, 256)  f32

    float* out  = (float*)d_out;
    float* loss = out;                               // first tuple element
    float* q    = out + (size_t)N_ROWS * K_CLUST;    // second tuple element (also q1 scratch)

    float* ws       = (float*)d_ws;
    float* Sval     = ws;                            // [0]
    float* partials = ws + 8;                        // 1024 block partials
    float* z2       = ws + 8 + 1024;                 // 16384
    float* w2       = z2 + N_ROWS;                   // 1024

    rowsq_kernel<<<N_ROWS / 8, 256, 0, stream>>>(comz, z2, N_ROWS);
    rowsq_kernel<<<K_CLUST / 8, 256, 0, stream>>>(weights, w2, K_CLUST);
    qgemm_kernel<<<dim3(K_CLUST / 128, N_ROWS / 128), 256, 0, stream>>>(
        comz, weights, z2, w2, q, partials);
    reduce_kernel<<<1, 256, 0, stream>>>(partials, Sval);
    finalize_kernel<<<4096, 256, 0, stream>>>(loss, q, Sval);
}